// GraphReasoningModel_79302276153731
// MI455X (gfx1250) — compile-verified
//
#include <hip/hip_runtime.h>
#include <hip/hip_bf16.h>

#define WAYS_  2
#define STEPS_ 3
#define Bn 16
#define Sn 32
#define Hn 768
#define En 200000
#define Tn 200000
#define Rn 500

typedef __attribute__((ext_vector_type(2))) float v2f;
typedef __attribute__((ext_vector_type(8))) float v8f;

// ---------------------------------------------------------------------------
// out(16 x N) = act( A(16 x K) @ W(K x N) + bias(N) )
// One 16x16 output tile per wave32 block using V_WMMA_F32_16X16X4_F32.
// A-frag layout (16x4 f32, 2 VGPRs): lanes 0-15 row M=lane hold K={k0,k0+1},
//   lanes 16-31 row M=lane-16 hold K={k0+2,k0+3}.
// B-frag layout (4x16 f32, 2 VGPRs): lanes 0-15 col N hold K rows {k0,k0+1},
//   lanes 16-31 hold K rows {k0+2,k0+3}.
// C/D layout: lane<16 -> col n0+lane rows 0..7 ; lane>=16 -> col rows 8..15.
// act: 0 = tanh, 1 = sigmoid.
// ---------------------------------------------------------------------------
__global__ __launch_bounds__(32) void gemm16_wmma(
    const float* __restrict__ A, const float* __restrict__ W,
    const float* __restrict__ bias, float* __restrict__ out,
    int K, int N, int act)
{
  const int lane = threadIdx.x;
  const int half = lane >> 4;     // 0 or 1
  const int l15  = lane & 15;
  const int n0   = blockIdx.x * 16;
  const int n    = n0 + l15;
  const bool nok = (n < N);
  const int nc   = nok ? n : (N - 1);   // clamped column (masked to 0 below)
  const int mA   = l15;                 // A row for this lane

  v8f acc = {};
  #pragma unroll 4
  for (int k0 = 0; k0 < K; k0 += 4) {
    v2f a, b;
    const int ka = k0 + 2 * half;
    a.x = A[mA * K + ka + 0];
    a.y = A[mA * K + ka + 1];
    float b0 = W[(size_t)(ka + 0) * N + nc];
    float b1 = W[(size_t)(ka + 1) * N + nc];
    b.x = nok ? b0 : 0.0f;              // cndmask select, EXEC stays full
    b.y = nok ? b1 : 0.0f;
    acc = __builtin_amdgcn_wmma_f32_16x16x4_f32(
        /*neg_a=*/false, a, /*neg_b=*/false, b,
        /*c_mod=*/(short)0, acc, /*reuse_a=*/false, /*reuse_b=*/false);
  }

  if (nok) {
    const float bv = bias[n];
    #pragma unroll
    for (int v = 0; v < 8; ++v) {
      const int m = half * 8 + v;
      float x = acc[v] + bv;
      float y = (act == 0) ? tanhf(x) : (1.0f / (1.0f + __expf(-x)));
      out[(size_t)m * N + n] = y;
    }
  }
}

// ---------------------------------------------------------------------------
// Per-batch attention: logits -> softmax -> mask -> renorm -> context vector.
// One wave32 per batch; lane == s (S == 32 == warpSize).
// ---------------------------------------------------------------------------
__global__ __launch_bounds__(32) void attn_ctx_kernel(
    const float* __restrict__ cq,    // (B,H)
    const float* __restrict__ qwh,   // (B,S,H)
    const float* __restrict__ amask, // (B,S)
    float* __restrict__ ctx)         // (B,H)
{
  const int b    = blockIdx.x;
  const int lane = threadIdx.x;                 // s index
  const float* qb  = qwh + (size_t)b * Sn * Hn;
  const float* cqb = cq  + (size_t)b * Hn;
  const float* row = qb  + (size_t)lane * Hn;

  float logit = 0.0f;
  for (int h = 0; h < Hn; ++h) logit += cqb[h] * row[h];

  float m = logit;
  for (int off = 16; off > 0; off >>= 1) m = fmaxf(m, __shfl_xor(m, off, 32));
  float e = __expf(logit - m);
  float esum = e;
  for (int off = 16; off > 0; off >>= 1) esum += __shfl_xor(esum, off, 32);
  float dist = (e / esum) * amask[b * Sn + lane];
  float dsum = dist;
  for (int off = 16; off > 0; off >>= 1) dsum += __shfl_xor(dsum, off, 32);
  dist = dist / (dsum + 1e-6f);

  for (int h = lane; h < Hn; h += 32) {
    float acc = 0.0f;
    #pragma unroll
    for (int s = 0; s < Sn; ++s) {
      float ds = __shfl(dist, s, 32);
      acc += ds * qb[(size_t)s * Hn + h];
    }
    ctx[(size_t)b * Hn + h] = acc;
  }
}

// ---------------------------------------------------------------------------
// hop_attn = softmax(q_emb @ hop_W[w] + hop_b[w]) over STEPS=3. One lane per b.
// ---------------------------------------------------------------------------
__global__ __launch_bounds__(32) void hop_attn_kernel(
    const float* __restrict__ qemb,  // (B,H)
    const float* __restrict__ hopW,  // (H,STEPS) slice for this way
    const float* __restrict__ hopB,  // (STEPS)
    float* __restrict__ hopA)        // (B,STEPS)
{
  const int b = threadIdx.x;
  if (b >= Bn) return;
  float l[STEPS_];
  #pragma unroll
  for (int t = 0; t < STEPS_; ++t) l[t] = hopB[t];
  const float* q = qemb + (size_t)b * Hn;
  for (int h = 0; h < Hn; ++h) {
    const float qv = q[h];
    #pragma unroll
    for (int t = 0; t < STEPS_; ++t) l[t] += qv * hopW[h * STEPS_ + t];
  }
  float m = fmaxf(l[0], fmaxf(l[1], l[2]));
  float s = 0.0f;
  #pragma unroll
  for (int t = 0; t < STEPS_; ++t) { l[t] = __expf(l[t] - m); s += l[t]; }
  #pragma unroll
  for (int t = 0; t < STEPS_; ++t) hopA[b * STEPS_ + t] = l[t] / s;
}

// ---------------------------------------------------------------------------
// last_e = heads ; score = 0   (grid-stride)
// ---------------------------------------------------------------------------
__global__ __launch_bounds__(256) void init_way_kernel(
    const float* __restrict__ heads, float* __restrict__ last_e,
    float* __restrict__ score, size_t n)
{
  size_t stride = (size_t)gridDim.x * blockDim.x;
  for (size_t i = (size_t)blockIdx.x * blockDim.x + threadIdx.x; i < n; i += stride) {
    last_e[i] = heads[i];
    score[i]  = 0.0f;
  }
}

__global__ __launch_bounds__(256) void zero_kernel(float* __restrict__ p, size_t n)
{
  size_t stride = (size_t)gridDim.x * blockDim.x;
  for (size_t i = (size_t)blockIdx.x * blockDim.x + threadIdx.x; i < n; i += stride)
    p[i] = 0.0f;
}

// ---------------------------------------------------------------------------
// propagate: new_e[b,obj] += last_e[b,sub] * rel_dist[b,rel]
// All arrays are L2-resident (12.8 MB each); limit is atomic throughput.
// ---------------------------------------------------------------------------
__global__ __launch_bounds__(256) void propagate_kernel(
    const int* __restrict__ triples,   // (B,T,3)
    const float* __restrict__ last_e,  // (B,E)
    const float* __restrict__ reld,    // (B,R)
    float* __restrict__ new_e)         // (B,E)
{
  const int i = blockIdx.x * 256 + threadIdx.x;
  const int b = blockIdx.y;
  if (i >= Tn) return;
  const int* tr = triples + ((size_t)b * Tn + i) * 3;
  const int sub = tr[0];
  const int rel = tr[1];
  const int obj = tr[2];
  const float v = last_e[(size_t)b * En + sub] * reld[b * Rn + rel];
  atomicAdd(new_e + (size_t)b * En + obj, v);
}

// ---------------------------------------------------------------------------
// v = min(new_e, 1) for v>1 (ref: v / max(v,1)); last_e = v;
// score += hop_attn[b,t] * v
// ---------------------------------------------------------------------------
__global__ __launch_bounds__(256) void norm_accum_kernel(
    const float* __restrict__ new_e, float* __restrict__ last_e,
    float* __restrict__ score, const float* __restrict__ hopA, int t)
{
  const int e = blockIdx.x * 256 + threadIdx.x;
  const int b = blockIdx.y;
  if (e >= En) return;
  const size_t idx = (size_t)b * En + e;
  float v = new_e[idx];
  v = (v > 1.0f) ? 1.0f : v;     // v / max(v,1)
  last_e[idx] = v;
  score[idx] += hopA[b * STEPS_ + t] * v;
}

__global__ __launch_bounds__(256) void final_mul_kernel(
    float* __restrict__ out, const float* __restrict__ scoreB, size_t n)
{
  size_t stride = (size_t)gridDim.x * blockDim.x;
  for (size_t i = (size_t)blockIdx.x * blockDim.x + threadIdx.x; i < n; i += stride)
    out[i] *= scoreB[i];
}

// ---------------------------------------------------------------------------
extern "C" void kernel_launch(void* const* d_in, const int* in_sizes, int n_in,
                              void* d_out, int out_size, void* d_ws, size_t ws_size,
                              hipStream_t stream) {
  const float* heads = (const float*)d_in[0];
  const float* qemb  = (const float*)d_in[1];
  const float* qwh   = (const float*)d_in[2];
  const float* amask = (const float*)d_in[3];
  const float* stepW = (const float*)d_in[4];   // (W,STEPS,H,H)
  const float* stepB = (const float*)d_in[5];   // (W,STEPS,H)
  const float* relW  = (const float*)d_in[6];   // (W,H,R)
  const float* relB  = (const float*)d_in[7];   // (W,R)
  const float* hopW  = (const float*)d_in[8];   // (W,H,STEPS)
  const float* hopB  = (const float*)d_in[9];   // (W,STEPS)
  const int*   trip  = (const int*)d_in[10];    // (B,T,3)

  float* ws = (float*)d_ws;
  const size_t BE = (size_t)Bn * En;            // 3.2M floats
  float* last_e = ws;                           // 12.8 MB
  float* new_e  = ws + BE;                      // 12.8 MB
  float* scoreB = ws + 2 * BE;                  // 12.8 MB
  float* cq     = ws + 3 * BE;                  // B*H
  float* ctx    = cq  + (size_t)Bn * Hn;        // B*H
  float* reld   = ctx + (size_t)Bn * Hn;        // B*R
  float* hopA   = reld + (size_t)Bn * Rn;       // B*STEPS

  const dim3 beGrid((En + 255) / 256, Bn);
  const dim3 tGrid((Tn + 255) / 256, Bn);

  for (int w = 0; w < WAYS_; ++w) {
    float* score = (w == 0) ? (float*)d_out : scoreB;
    init_way_kernel<<<2048, 256, 0, stream>>>(heads, last_e, score, BE);
    hop_attn_kernel<<<1, 32, 0, stream>>>(
        qemb, hopW + (size_t)w * Hn * STEPS_, hopB + w * STEPS_, hopA);
    for (int t = 0; t < STEPS_; ++t) {
      // cq_t = tanh(q_emb @ step_W[w,t] + step_b[w,t])   16x768x768, WMMA f32
      gemm16_wmma<<<Hn / 16, 32, 0, stream>>>(
          qemb, stepW + (size_t)(w * STEPS_ + t) * Hn * Hn,
          stepB + (size_t)(w * STEPS_ + t) * Hn, cq, Hn, Hn, /*tanh*/0);
      // softmax attention + context
      attn_ctx_kernel<<<Bn, 32, 0, stream>>>(cq, qwh, amask, ctx);
      // rel_dist = sigmoid(ctx @ rel_W[w] + rel_b[w])    16x768x500, WMMA f32
      gemm16_wmma<<<(Rn + 15) / 16, 32, 0, stream>>>(
          ctx, relW + (size_t)w * Hn * Rn, relB + (size_t)w * Rn,
          reld, Hn, Rn, /*sigmoid*/1);
      // sparse propagate + clamp-normalize + score accumulation
      zero_kernel<<<2048, 256, 0, stream>>>(new_e, BE);
      propagate_kernel<<<tGrid, 256, 0, stream>>>(trip, last_e, reld, new_e);
      norm_accum_kernel<<<beGrid, 256, 0, stream>>>(new_e, last_e, score, hopA, t);
    }
  }
  // out = score_way0 * score_way1
  final_mul_kernel<<<2048, 256, 0, stream>>>((float*)d_out, scoreB, BE);
}